// ChunkedValueCrossAttn_84430467105062
// MI455X (gfx1250) — compile-verified
//
#include <hip/hip_runtime.h>
#include <hip/hip_bf16.h>

// ---- Problem constants (from the reference) -------------------------------
// x: [2,64,1024,1024] (DEAD: softmax over size-1 axis => attn == 1)
// y[b,c,h,w] = (Wo @ (Wv @ context[b]))[c] + bo[c]   -- constant per (b,c)
// Output: 2*64*1024*1024 f32 = 512 MiB  -> pure store-bandwidth problem
// (~536 MB / 23.3 TB/s ~= 23 us; compute is negligible, x must NOT be read).

typedef __attribute__((ext_vector_type(2))) float v2f;
typedef __attribute__((ext_vector_type(4))) float v4f;
typedef __attribute__((ext_vector_type(8))) float v8f;

// ---------------------------------------------------------------------------
// Kernel 1: one wave32 computes vals[b*64+c] = (Wo @ Wv @ ctx[b])[c] + bo[c]
// using V_WMMA_F32_16X16X4_F32 (full f32 precision).
//
// WMMA f32 16x16x4 fragment layouts (ISA 7.12.2):
//   A (16x4): lane L (m=L&15, half=L>>4): a.x=A[m][k0+2*half], a.y=A[m][k0+2*half+1]
//   B (4x16): lane L (n=L&15, half=L>>4): b.x=B[k0+2*half][n], b.y=B[k0+2*half+1][n]
//   C/D (16x16): lane L (n=L&15, half=L>>4): d[r] = D[half*8+r][n], r=0..7
// ---------------------------------------------------------------------------
__global__ __launch_bounds__(32) void cvca_compute_vals(
    const float* __restrict__ ctx,   // [2,16]
    const float* __restrict__ Wv,    // [32,16]
    const float* __restrict__ Wo,    // [64,32]
    const float* __restrict__ bo,    // [64]
    float* __restrict__ vals)        // [128] = [b*64 + c]
{
    __shared__ float ldsV[32 * 16];  // Vmat[o][n]; cols 0..1 = batch, rest pad

    const int lane = threadIdx.x;    // 0..31 (wave32)
    const int half = lane >> 4;      // 0 or 1
    const int mn   = lane & 15;      // m (for A/D rows) or n (for B/D cols)

    // ---- Stage 1: Vmat(32x16) = Wv(32x16) x CtxT(16x16; cols 0..1 valid) ---
    #pragma unroll
    for (int t = 0; t < 2; ++t) {
        const int o0 = t * 16;
        v8f acc = {};
        #pragma unroll
        for (int k0 = 0; k0 < 16; k0 += 4) {
            const int kb = k0 + half * 2;
            v2f a, b;
            a.x = Wv[(o0 + mn) * 16 + kb];
            a.y = Wv[(o0 + mn) * 16 + kb + 1];
            // B[k][n] = ctx[n][k] for n<2, else 0 (zero-pad columns 2..15)
            float bx = 0.0f, by = 0.0f;
            if (mn < 2) {
                bx = ctx[mn * 16 + kb];
                by = ctx[mn * 16 + kb + 1];
            }
            b.x = bx; b.y = by;
            acc = __builtin_amdgcn_wmma_f32_16x16x4_f32(
                false, a, false, b, (short)0, acc, false, false);
        }
        // Scatter D tile into LDS: row = o0 + half*8 + r, col = mn
        #pragma unroll
        for (int r = 0; r < 8; ++r)
            ldsV[(o0 + half * 8 + r) * 16 + mn] = acc[r];
    }
    __syncthreads();

    // ---- Stage 2: Y(64x16) = Wo(64x32) x Vmat(32x16); cols 0..1 valid ------
    #pragma unroll
    for (int t = 0; t < 4; ++t) {
        const int c0 = t * 16;
        v8f acc = {};
        #pragma unroll
        for (int k0 = 0; k0 < 32; k0 += 4) {
            const int kb = k0 + half * 2;
            v2f a, b;
            a.x = Wo[(c0 + mn) * 32 + kb];
            a.y = Wo[(c0 + mn) * 32 + kb + 1];
            b.x = ldsV[kb * 16 + mn];
            b.y = ldsV[(kb + 1) * 16 + mn];
            acc = __builtin_amdgcn_wmma_f32_16x16x4_f32(
                false, a, false, b, (short)0, acc, false, false);
        }
        // Columns n = 0,1 carry the two batch entries.
        if (mn < 2) {
            const int bidx  = mn;
            const int mbase = c0 + half * 8;
            #pragma unroll
            for (int r = 0; r < 8; ++r) {
                const int c = mbase + r;
                vals[bidx * 64 + c] = acc[r] + bo[c];
            }
        }
    }
}

// ---------------------------------------------------------------------------
// Kernel 2: broadcast fill, store-bandwidth bound (~536 MB @ 23.3 TB/s).
// Each (b,c) plane = 1<<20 floats = 1<<18 float4s, covered by 64 blocks.
// plane index is block-uniform -> vals[plane] becomes a scalar load.
// 16 x 128-bit non-temporal stores per thread (GLOBAL_STORE_B128, TH=NT:
// output (512 MiB) exceeds the 192 MB L2 and is write-once).
// ---------------------------------------------------------------------------
__global__ __launch_bounds__(256) void cvca_broadcast_fill(
    const float* __restrict__ vals,  // [128]
    v4f* __restrict__ out)           // [128 * (1<<18)] 16B vectors
{
    const int plane      = (int)(blockIdx.x >> 6);   // 64 blocks per plane
    const int blkInPlane = (int)(blockIdx.x & 63);
    const float v = vals[plane];                     // block-uniform (s_load)
    const v4f q = {v, v, v, v};

    // Block covers 256 threads * 16 float4s = 4096 float4s of the plane.
    size_t base = (size_t)plane * (1u << 18)
                + (size_t)blkInPlane * 4096u
                + (size_t)threadIdx.x;
    #pragma unroll
    for (int i = 0; i < 16; ++i)
        __builtin_nontemporal_store(q, &out[base + (size_t)i * 256u]);
}

// ---------------------------------------------------------------------------
extern "C" void kernel_launch(void* const* d_in, const int* in_sizes, int n_in,
                              void* d_out, int out_size, void* d_ws, size_t ws_size,
                              hipStream_t stream) {
    // Inputs per setup_inputs() order:
    // 0: x (unused), 1: context, 2: Wq (unused), 3: Wk (unused),
    // 4: Wv, 5: Wo, 6: bo
    const float* ctx = (const float*)d_in[1];
    const float* Wv  = (const float*)d_in[4];
    const float* Wo  = (const float*)d_in[5];
    const float* bo  = (const float*)d_in[6];

    float* vals = (float*)d_ws;           // 128 floats of scratch
    v4f*   out  = (v4f*)d_out;            // 2*64*1024*1024 f32

    cvca_compute_vals<<<1, 32, 0, stream>>>(ctx, Wv, Wo, bo, vals);

    // 128 planes * 64 blocks/plane = 8192 blocks of 256 threads.
    cvca_broadcast_fill<<<8192, 256, 0, stream>>>(vals, out);
}